// GCN_86732569575636
// MI455X (gfx1250) — compile-verified
//
#include <hip/hip_runtime.h>

typedef __attribute__((ext_vector_type(2))) float v2f;
typedef __attribute__((ext_vector_type(8))) float v8f;
typedef __attribute__((ext_vector_type(4))) unsigned int v4u;
typedef __attribute__((ext_vector_type(8))) int v8i;
typedef __attribute__((ext_vector_type(4))) int v4i;

// ---------------- utility ----------------
__global__ void fill_f32(float* __restrict__ p, float v, int n) {
    int i = blockIdx.x * blockDim.x + threadIdx.x;
    if (i < n) p[i] = v;
}

// deg[dst]++ per edge (deg pre-initialized to 1.0 for the self-loop)
__global__ void degree_k(const int* __restrict__ ei, float* __restrict__ deg, int E) {
    int e = blockIdx.x * blockDim.x + threadIdx.x;
    if (e < E) atomicAdd(&deg[ei[E + e]], 1.0f);
}

__global__ void dinv_k(const float* __restrict__ deg, float* __restrict__ dinv, int N) {
    int i = blockIdx.x * blockDim.x + threadIdx.x;
    if (i < N) {
        float d = deg[i];
        dinv[i] = d > 0.0f ? rsqrtf(d) : 0.0f;
    }
}

// Layer-1 aggregation on RAW 3-dim features: agg1[dst] += norm * x[src]
// (exploits A(xW) = (Ax)W : 12 B/edge gathered instead of 512 B/edge)
__global__ void agg1_k(const int* __restrict__ ei, const float* __restrict__ x,
                       const float* __restrict__ dinv, float* __restrict__ agg1, int E) {
    int e = blockIdx.x * blockDim.x + threadIdx.x;
    if (e >= E) return;
    int s = ei[e], d = ei[E + e];
    float w = dinv[s] * dinv[d];
    atomicAdd(&agg1[d * 4 + 0], w * x[s * 3 + 0]);
    atomicAdd(&agg1[d * 4 + 1], w * x[s * 3 + 1]);
    atomicAdd(&agg1[d * 4 + 2], w * x[s * 3 + 2]);
}

// h = relu( (agg1 + dinv^2 * x) @ W1 + b1 )  via V_WMMA_F32_16X16X4_F32
// one wave per (16-row tile, 16-col tile); K = 4 (3 features zero-padded)
template <int HT>
__global__ void gemm1_wmma(const float* __restrict__ agg1, const float* __restrict__ x,
                           const float* __restrict__ dinv, const float* __restrict__ W1,
                           const float* __restrict__ b1, float* __restrict__ h,
                           int N, int Hrt) {
    const int H = (HT > 0) ? HT : Hrt;
    int wid  = (blockIdx.x * blockDim.x + threadIdx.x) >> 5;
    int lane = threadIdx.x & 31;
    int ctn  = H >> 4;
    int tiles = ((N + 15) >> 4) * ctn;
    if (wid >= tiles) return;                 // wave-uniform: EXEC all-1s at WMMA
    int rt = wid / ctn, ct = wid % ctn;
    int l  = lane & 15;
    bool hi = lane >= 16;
    bool full = (rt * 16 + 16) <= N;

    int m = rt * 16 + l;
    v2f a = {0.0f, 0.0f};
    if (full || m < N) {
        float di = dinv[m];
        float w  = di * di;                   // folded self-loop
        if (!hi) {
            v2f g = *(const v2f*)(agg1 + (size_t)m * 4);      // b64 load (K=0,1)
            a.x = g.x + w * x[m * 3 + 0];
            a.y = g.y + w * x[m * 3 + 1];
        } else {
            a.x = agg1[(size_t)m * 4 + 2] + w * x[m * 3 + 2]; // K=2
            a.y = 0.0f;                                        // K=3 pad
        }
    }
    int n = ct * 16 + l;
    v2f b = {0.0f, 0.0f};
    if (!hi) { b.x = W1[0 * H + n]; b.y = W1[1 * H + n]; }
    else     { b.x = W1[2 * H + n]; b.y = 0.0f; }

    v8f c = {};
    c = __builtin_amdgcn_wmma_f32_16x16x4_f32(false, a, false, b, (short)0, c, false, false);

    float bias = b1[n];
    int r0 = rt * 16 + (hi ? 8 : 0);
    float* hp = h + (size_t)r0 * H + n;       // constant-stride stores (H templated)
    if (full) {
#pragma unroll
        for (int j = 0; j < 8; ++j) {
            float v = c[j] + bias;
            hp[(size_t)j * H] = v > 0.0f ? v : 0.0f;
        }
    } else {
#pragma unroll
        for (int j = 0; j < 8; ++j) {
            if (r0 + j < N) {
                float v = c[j] + bias;
                hp[(size_t)j * H] = v > 0.0f ? v : 0.0f;
            }
        }
    }
}

// p = h @ W2  (N x 128 @ 128 x 10, output cols padded to 16).
// W2 (128 x 10 f32 tile) is staged into LDS by the Tensor Data Mover:
// one tensor_load_to_lds per block, s_wait_tensorcnt 0, then 32 chained WMMAs.
template <int HT>
__global__ void gemm2_wmma(const float* __restrict__ h, const float* __restrict__ W2,
                           float* __restrict__ p, int N, int Hrt, int C) {
    __shared__ float sW2[128 * 10];
    const int H = (HT > 0) ? HT : Hrt;

    if (threadIdx.x == 0) {
        unsigned int lds = (unsigned int)(uintptr_t)(void*)sW2;
        unsigned long long ga = (unsigned long long)(uintptr_t)W2;
        // D# group 0: count=1 | lds_addr | global_addr(57b) | type=2
        v4u g0 = { 1u,
                   lds,
                   (unsigned int)(ga & 0xFFFFFFFFu),
                   (unsigned int)((ga >> 32) & 0x01FFFFFFu) | 0x80000000u };
        // D# group 1: data_size=4B; tensor_dim0=C, tensor_dim1=H;
        //             tile_dim0=C, tile_dim1=H, tile_dim2=0; dim0_stride=C
        v8i g1 = { (int)(2u << 16),
                   (int)((unsigned int)C << 16),
                   (int)((unsigned int)H << 16),
                   (int)((unsigned int)C << 16),
                   (int)(unsigned int)H,
                   (int)C,
                   0, 0 };
        v4i g2 = {0, 0, 0, 0};
        v4i g3 = {0, 0, 0, 0};
        v8i g4 = {0, 0, 0, 0, 0, 0, 0, 0};
        __builtin_amdgcn_tensor_load_to_lds(g0, g1, g2, g3, g4, 0);
        __builtin_amdgcn_s_wait_tensorcnt(0);
    }
    __syncthreads();

    int wid  = (blockIdx.x * blockDim.x + threadIdx.x) >> 5;
    int lane = threadIdx.x & 31;
    int l    = lane & 15;
    bool hi  = lane >= 16;
    int tiles = (N + 15) >> 4;
    if (wid >= tiles) return;
    int m = wid * 16 + l;
    bool full = (wid * 16 + 16) <= N;

    v8f c = {};
#pragma unroll
    for (int k0 = 0; k0 < ((HT > 0) ? HT : 1024); k0 += 4) {
        if (HT == 0 && k0 >= H) break;
        v2f a = {0.0f, 0.0f};
        if (full || m < N)
            a = *(const v2f*)(h + (size_t)m * H + k0 + (hi ? 2 : 0));  // b64 load
        v2f b = {0.0f, 0.0f};
        if (l < C) {
            if (!hi) { b.x = sW2[(k0 + 0) * C + l]; b.y = sW2[(k0 + 1) * C + l]; }
            else     { b.x = sW2[(k0 + 2) * C + l]; b.y = sW2[(k0 + 3) * C + l]; }
        }
        c = __builtin_amdgcn_wmma_f32_16x16x4_f32(false, a, false, b, (short)0, c, false, false);
    }

    float* pp = p + (size_t)(wid * 16 + (hi ? 8 : 0)) * 16 + l;
    if (full) {
#pragma unroll
        for (int j = 0; j < 8; ++j) pp[j * 16] = c[j];
    } else {
        int r0 = wid * 16 + (hi ? 8 : 0);
#pragma unroll
        for (int j = 0; j < 8; ++j)
            if (r0 + j < N) pp[j * 16] = c[j];
    }
}

// Layer-2 aggregation on the 10-dim transformed features: agg2[dst] += norm * p[src]
__global__ void agg2_k(const int* __restrict__ ei, const float* __restrict__ p,
                       const float* __restrict__ dinv, float* __restrict__ agg2,
                       int E, int C) {
    int e = blockIdx.x * blockDim.x + threadIdx.x;
    if (e >= E) return;
    int s = ei[e], d = ei[E + e];
    float w = dinv[s] * dinv[d];
    const float4* ps4 = (const float4*)(p + (size_t)s * 16);   // b128 gathers
    float4 q0 = ps4[0], q1 = ps4[1], q2 = ps4[2];
    float pv[12] = { q0.x, q0.y, q0.z, q0.w, q1.x, q1.y, q1.z, q1.w,
                     q2.x, q2.y, q2.z, q2.w };
    float* ad = agg2 + (size_t)d * 16;
    for (int c = 0; c < C; ++c) atomicAdd(&ad[c], w * pv[c]);
}

// self-loop fold + bias + segment-sum pool (+ count)
__global__ void pool_k(const float* __restrict__ agg2, const float* __restrict__ p,
                       const float* __restrict__ dinv, const float* __restrict__ b2,
                       const int* __restrict__ batch, float* __restrict__ pooled,
                       float* __restrict__ cnt, int N, int C) {
    int i = blockIdx.x * blockDim.x + threadIdx.x;
    if (i >= N) return;
    float di = dinv[i];
    float w  = di * di;
    int g = batch[i];
    const float4* a4 = (const float4*)(agg2 + (size_t)i * 16);
    const float4* p4 = (const float4*)(p + (size_t)i * 16);
    float4 a0 = a4[0], a1 = a4[1], a2 = a4[2];
    float4 q0 = p4[0], q1 = p4[1], q2 = p4[2];
    float av[12] = { a0.x, a0.y, a0.z, a0.w, a1.x, a1.y, a1.z, a1.w,
                     a2.x, a2.y, a2.z, a2.w };
    float pv[12] = { q0.x, q0.y, q0.z, q0.w, q1.x, q1.y, q1.z, q1.w,
                     q2.x, q2.y, q2.z, q2.w };
    float* pg = pooled + (size_t)g * 16;
    for (int c = 0; c < C; ++c)
        atomicAdd(&pg[c], av[c] + w * pv[c] + b2[c]);
    atomicAdd(&cnt[g], 1.0f);
}

// mean + log_softmax epilogue, one thread per graph
__global__ void lsm_k(const float* __restrict__ pooled, const float* __restrict__ cnt,
                      float* __restrict__ out, int G, int C) {
    int g = blockIdx.x * blockDim.x + threadIdx.x;
    if (g >= G) return;
    float inv = 1.0f / fmaxf(cnt[g], 1.0f);
    float v[16];
    float mx = -3.4e38f;
    for (int c = 0; c < C; ++c) { v[c] = pooled[(size_t)g * 16 + c] * inv; mx = fmaxf(mx, v[c]); }
    float s = 0.0f;
    for (int c = 0; c < C; ++c) s += expf(v[c] - mx);
    float lse = logf(s) + mx;
    for (int c = 0; c < C; ++c) out[(size_t)g * C + c] = v[c] - lse;
}

extern "C" void kernel_launch(void* const* d_in, const int* in_sizes, int n_in,
                              void* d_out, int out_size, void* d_ws, size_t ws_size,
                              hipStream_t stream) {
    const float* x  = (const float*)d_in[0];
    const float* W1 = (const float*)d_in[1];
    const float* b1 = (const float*)d_in[2];
    const float* W2 = (const float*)d_in[3];
    const float* b2 = (const float*)d_in[4];
    const int*   ei = (const int*)d_in[5];     // (2,E): row 0 = src, row 1 = dst
    const int*   bt = (const int*)d_in[6];

    const int N = in_sizes[0] / 3;
    const int H = in_sizes[2];
    const int C = in_sizes[4];
    const int E = in_sizes[5] / 2;
    const int G = out_size / C;

    float* ws     = (float*)d_ws;
    float* deg    = ws;                 ws += N;
    float* dinv   = ws;                 ws += N;
    float* agg1   = ws;                 ws += (size_t)4 * N;
    float* agg2   = ws;                 ws += (size_t)16 * N;
    float* pooled = ws;                 ws += (size_t)16 * G;
    float* cnt    = ws;                 ws += G;
    float* h      = ws;                 ws += (size_t)H * N;
    float* p      = ws;                 /* 16*N */

    const int B = 256;
    // zero agg1 | agg2 | pooled | cnt (contiguous), deg = 1.0 (self-loop)
    int zc = 4 * N + 16 * N + 16 * G + G;
    fill_f32<<<(zc + B - 1) / B, B, 0, stream>>>(agg1, 0.0f, zc);
    fill_f32<<<(N + B - 1) / B, B, 0, stream>>>(deg, 1.0f, N);

    degree_k<<<(E + B - 1) / B, B, 0, stream>>>(ei, deg, E);
    dinv_k<<<(N + B - 1) / B, B, 0, stream>>>(deg, dinv, N);
    agg1_k<<<(E + B - 1) / B, B, 0, stream>>>(ei, x, dinv, agg1, E);

    long long rtiles = (N + 15) / 16;
    {
        long long thr = rtiles * (H / 16) * 32;     // one wave per 16x16 output tile
        if (H == 128)
            gemm1_wmma<128><<<(thr + B - 1) / B, B, 0, stream>>>(agg1, x, dinv, W1, b1, h, N, H);
        else
            gemm1_wmma<0><<<(thr + B - 1) / B, B, 0, stream>>>(agg1, x, dinv, W1, b1, h, N, H);
    }
    {
        long long thr = rtiles * 32;                // one wave per 16-row tile
        if (H == 128)
            gemm2_wmma<128><<<(thr + B - 1) / B, B, 0, stream>>>(h, W2, p, N, H, C);
        else
            gemm2_wmma<0><<<(thr + B - 1) / B, B, 0, stream>>>(h, W2, p, N, H, C);
    }

    agg2_k<<<(E + B - 1) / B, B, 0, stream>>>(ei, p, dinv, agg2, E, C);
    pool_k<<<(N + B - 1) / B, B, 0, stream>>>(agg2, p, dinv, b2, bt, pooled, cnt, N, C);
    lsm_k<<<(G + B - 1) / B, B, 0, stream>>>(pooled, cnt, (float*)d_out, G, C);
}